// SupConLoss2_72756745994686
// MI455X (gfx1250) — compile-verified
//
#include <hip/hip_runtime.h>
#include <hip/hip_bf16.h>
#include <math.h>

typedef __attribute__((ext_vector_type(2))) float v2f;
typedef __attribute__((ext_vector_type(8))) float v8f;

#define BSZ    4096
#define NROWS  8192                    // bsz * n_views
#define DIM    128
#define NT     (NROWS / 16)            // 512 column tiles
#define INVT   (1.0f / 0.07f)
#define C_EXP2 (1.4426950408889634f / 0.07f)   // log2(e)/T
#define LN2F   0.6931471805599453f

// row i of the view-major stacked matrix: b = i % 4096, v = i / 4096
// features[b, v, k] lives at ((b*2 + v) * 128 + k)
__device__ __forceinline__ int feat_off(int i) {
  return (((i & (BSZ - 1)) << 1) + (i >> 12)) * DIM;
}

__device__ __forceinline__ float hmax16(float v) {
  v = fmaxf(v, __shfl_xor(v, 1, 32));
  v = fmaxf(v, __shfl_xor(v, 2, 32));
  v = fmaxf(v, __shfl_xor(v, 4, 32));
  v = fmaxf(v, __shfl_xor(v, 8, 32));
  return v;
}
__device__ __forceinline__ float hsum16(float v) {
  v += __shfl_xor(v, 1, 32);
  v += __shfl_xor(v, 2, 32);
  v += __shfl_xor(v, 4, 32);
  v += __shfl_xor(v, 8, 32);
  return v;
}

// one 16x16 tile of A·A^T via 32 K-steps of V_WMMA_F32_16X16X4_F32
__device__ __forceinline__ v8f tile_mm(const float* __restrict__ F,
                                       const v2f* __restrict__ afrag,
                                       int col_row, int khalf) {
  const float2* bp = (const float2*)(F + feat_off(col_row) + khalf * 2);
  v8f acc = {};
#pragma unroll
  for (int t = 0; t < 32; ++t) {
    float2 b = bp[t * 2];
    v2f bf; bf.x = b.x; bf.y = b.y;
    acc = __builtin_amdgcn_wmma_f32_16x16x4_f32(false, afrag[t], false, bf,
                                                (short)0, acc, false, false);
  }
  return acc;
}

__global__ __launch_bounds__(128)
void supcon_rows_kernel(const float* __restrict__ F,
                        const int* __restrict__ labels,
                        float* __restrict__ rowloss) {
  __shared__ int slab[BSZ];
  for (int i = threadIdx.x; i < BSZ; i += 128) slab[i] = labels[i];
  __syncthreads();

  const int lane    = threadIdx.x & 31;
  const int wave    = threadIdx.x >> 5;
  const int n16     = lane & 15;   // N (col) within tile; also M for A-frag
  const int khalf   = lane >> 4;   // 0/1 -> K offset 0 or 2
  const int rowbase = blockIdx.x * 64 + wave * 16;

  // ---- hoist A fragments: 16 rows x 128 K, 32 k-steps of 4 ----
  const int arow = rowbase + n16;
  const float2* ap = (const float2*)(F + feat_off(arow) + khalf * 2);
  v2f afrag[32];
#pragma unroll
  for (int t = 0; t < 32; ++t) {
    float2 a = ap[t * 2];
    afrag[t].x = a.x; afrag[t].y = a.y;
  }

  // per-lane metadata: this lane's accumulator slot r covers row r + 8*khalf
  int rlab[8];
#pragma unroll
  for (int r = 0; r < 8; ++r)
    rlab[r] = slab[(rowbase + r + 8 * khalf) & (BSZ - 1)];

  // ========== pass 1: row max of raw dots (diag included, monotone in 1/T) ==
  float rmax[8];
#pragma unroll
  for (int r = 0; r < 8; ++r) rmax[r] = -3.4e38f;

  for (int ct = 0; ct < NT; ++ct) {
    v8f acc = tile_mm(F, afrag, ct * 16 + n16, khalf);
#pragma unroll
    for (int r = 0; r < 8; ++r) rmax[r] = fmaxf(rmax[r], acc[r]);
  }
  float m2[8];
#pragma unroll
  for (int r = 0; r < 8; ++r) {
    rmax[r] = hmax16(rmax[r]);
    m2[r]   = rmax[r] * C_EXP2;        // max in exp2 domain
  }

  // ========== pass 2: exp-sum (excl. self), positive dot sum, count =========
  float s[8], sp[8], cnt[8];
#pragma unroll
  for (int r = 0; r < 8; ++r) { s[r] = 0.f; sp[r] = 0.f; cnt[r] = 0.f; }

  const int dt = rowbase >> 4;         // the only tile that touches the diagonal

  auto body = [&](int ct, bool diag) {
    const int clab = slab[(ct * 16 + n16) & (BSZ - 1)];
    v8f acc = tile_mm(F, afrag, ct * 16 + n16, khalf);
#pragma unroll
    for (int r = 0; r < 8; ++r) {
      float d = acc[r];
      float e = __builtin_amdgcn_exp2f(fmaf(d, C_EXP2, -m2[r]));
      bool pos = (clab == rlab[r]);
      if (diag) {
        bool self = (n16 == r + 8 * khalf);
        e   = self ? 0.f : e;
        pos = pos && !self;
      }
      s[r]   += e;
      sp[r]  += pos ? d : 0.f;
      cnt[r] += pos ? 1.f : 0.f;
    }
  };

  for (int ct = 0; ct < dt; ++ct)       body(ct, false);
  body(dt, true);
  for (int ct = dt + 1; ct < NT; ++ct)  body(ct, false);

  // ===== per-row finalize: -( (sp/T)/cnt - m/T - ln(sum_exp) ) ==============
#pragma unroll
  for (int r = 0; r < 8; ++r) {
    float ssum  = hsum16(s[r]);
    float spsum = hsum16(sp[r]);
    float csum  = hsum16(cnt[r]);
    if (n16 == 0) {
      float mlpp = (spsum * INVT) / csum - rmax[r] * INVT
                 - __builtin_amdgcn_logf(ssum) * LN2F;   // v_log_f32 = log2
      rowloss[rowbase + r + 8 * khalf] = -mlpp;
    }
  }
}

__global__ __launch_bounds__(256)
void supcon_reduce_kernel(const float* __restrict__ rowloss,
                          float* __restrict__ out) {
  float v = 0.f;
  for (int i = threadIdx.x; i < NROWS; i += 256) v += rowloss[i];
  v += __shfl_xor(v, 1, 32);
  v += __shfl_xor(v, 2, 32);
  v += __shfl_xor(v, 4, 32);
  v += __shfl_xor(v, 8, 32);
  v += __shfl_xor(v, 16, 32);
  __shared__ float wsum[8];
  if ((threadIdx.x & 31) == 0) wsum[threadIdx.x >> 5] = v;
  __syncthreads();
  if (threadIdx.x == 0) {
    float t = 0.f;
#pragma unroll
    for (int i = 0; i < 8; ++i) t += wsum[i];
    out[0] = t / (float)NROWS;
  }
}

extern "C" void kernel_launch(void* const* d_in, const int* in_sizes, int n_in,
                              void* d_out, int out_size, void* d_ws, size_t ws_size,
                              hipStream_t stream) {
  const float* features = (const float*)d_in[0];   // [4096, 2, 128] f32
  const int*   labels   = (const int*)d_in[1];     // [4096] int32
  float* rowloss = (float*)d_ws;                   // 8192 floats scratch
  float* out     = (float*)d_out;                  // scalar f32

  supcon_rows_kernel<<<NROWS / 64, 128, 0, stream>>>(features, labels, rowloss);
  supcon_reduce_kernel<<<1, 256, 0, stream>>>(rowloss, out);
}